// EfficientMemorySoftmax_58248346468923
// MI455X (gfx1250) — compile-verified
//
#include <hip/hip_runtime.h>
#include <hip/hip_bf16.h>
#include <stdint.h>

#define ROW    2048          // elements per softmax row (fixed by reference shape)
#define TPB    256           // 8 wave32s per block
#define WAVES  (TPB / 32)
#define BLOCKS 4096          // persistent blocks; rows / BLOCKS = 8 rows each

#if __has_builtin(__builtin_amdgcn_exp2f)
#define EXP2F(x) __builtin_amdgcn_exp2f(x)
#else
#define EXP2F(x) exp2f(x)
#endif

#if __has_builtin(__builtin_amdgcn_rcpf)
#define RCPF(x) __builtin_amdgcn_rcpf(x)
#else
#define RCPF(x) (1.0f / (x))
#endif

__device__ __forceinline__ float wave_max32(float v) {
#pragma unroll
  for (int off = 16; off > 0; off >>= 1)
    v = fmaxf(v, __shfl_xor(v, off, 32));
  return v;
}

__device__ __forceinline__ float wave_sum32(float v) {
#pragma unroll
  for (int off = 16; off > 0; off >>= 1)
    v += __shfl_xor(v, off, 32);
  return v;
}

// Issue this thread's two 16B async copies of row r into the LDS buffer whose
// base byte-address is lds_base. Chunk layout: thread t owns chunks t and t+TPB,
// so per-wave global addresses are consecutive 16B -> fully coalesced b128s.
__device__ __forceinline__ void prefetch_row_async(const float* __restrict__ x,
                                                   int r, uint32_t lds_base, int t) {
  const char* rb = (const char*)(x + (size_t)r * ROW);
  uint64_t g0 = (uint64_t)(uintptr_t)(rb + (size_t)t * 16);
  uint64_t g1 = g0 + (uint64_t)TPB * 16;
  uint32_t l0 = lds_base + (uint32_t)t * 16u;
  uint32_t l1 = l0 + TPB * 16u;
  asm volatile("global_load_async_to_lds_b128 %0, %1, off"
               :: "v"(l0), "v"(g0) : "memory");
  asm volatile("global_load_async_to_lds_b128 %0, %1, off"
               :: "v"(l1), "v"(g1) : "memory");
}

__global__ __launch_bounds__(TPB)
void softmax2048_kernel(const float* __restrict__ x, float* __restrict__ out, int rows) {
  __shared__ float buf[2][ROW];   // 2 x 8KB double buffer
  __shared__ float wmax[WAVES];
  __shared__ float wsum[WAVES];

  const int t    = threadIdx.x;
  const int wid  = t >> 5;
  const int lane = t & 31;

  // LDS byte addresses of the two buffers (generic-pointer low 32 bits = LDS offset)
  const uint32_t lds0 = (uint32_t)(uintptr_t)(&buf[0][0]);
  const uint32_t lds1 = (uint32_t)(uintptr_t)(&buf[1][0]);

  int r = blockIdx.x;
  if (r >= rows) return;
  const int stride = gridDim.x;

  // Prologue: start fetching first row into buffer 0.
  prefetch_row_async(x, r, lds0, t);

  int cur = 0;
  for (; r < rows; r += stride) {
    const int nxt = r + stride;           // uniform across the block
    if (nxt < rows) {
      // Kick off next row into the other buffer, then wait only for the
      // current row's two async groups (async loads complete in order).
      prefetch_row_async(x, nxt, cur ? lds0 : lds1, t);
      asm volatile("s_wait_asynccnt 0x2" ::: "memory");
    } else {
      asm volatile("s_wait_asynccnt 0x0" ::: "memory");
    }
    __syncthreads();                      // all waves' chunks of row r are in LDS

    // ---- read this thread's 8 elements (2 x ds_read_b128) ----
    const float4* bp = (const float4*)(cur ? &buf[1][0] : &buf[0][0]);
    float4 a = bp[t];
    float4 b = bp[t + TPB];

    // ---- block max ----
    float m = fmaxf(fmaxf(fmaxf(a.x, a.y), fmaxf(a.z, a.w)),
                    fmaxf(fmaxf(b.x, b.y), fmaxf(b.z, b.w)));
    m = wave_max32(m);
    if (lane == 0) wmax[wid] = m;
    __syncthreads();
    float rm = wmax[0];
#pragma unroll
    for (int i = 1; i < WAVES; ++i) rm = fmaxf(rm, wmax[i]);

    // ---- exp (native v_exp_f32 via exp2) and block sum ----
    const float L2E = 1.4426950408889634f;
    float4 ea, eb;
    ea.x = EXP2F((a.x - rm) * L2E);
    ea.y = EXP2F((a.y - rm) * L2E);
    ea.z = EXP2F((a.z - rm) * L2E);
    ea.w = EXP2F((a.w - rm) * L2E);
    eb.x = EXP2F((b.x - rm) * L2E);
    eb.y = EXP2F((b.y - rm) * L2E);
    eb.z = EXP2F((b.z - rm) * L2E);
    eb.w = EXP2F((b.w - rm) * L2E);

    float s = ((ea.x + ea.y) + (ea.z + ea.w)) + ((eb.x + eb.y) + (eb.z + eb.w));
    s = wave_sum32(s);
    if (lane == 0) wsum[wid] = s;
    __syncthreads();
    float rs = wsum[0];
#pragma unroll
    for (int i = 1; i < WAVES; ++i) rs += wsum[i];
    const float inv = RCPF(rs);

    // ---- normalize in registers, 2 x global_store_b128 ----
    ea.x *= inv; ea.y *= inv; ea.z *= inv; ea.w *= inv;
    eb.x *= inv; eb.y *= inv; eb.z *= inv; eb.w *= inv;

    float4* op = (float4*)(out + (size_t)r * ROW);
    op[t]       = ea;
    op[t + TPB] = eb;

    cur ^= 1;
  }
}

extern "C" void kernel_launch(void* const* d_in, const int* in_sizes, int n_in,
                              void* d_out, int out_size, void* d_ws, size_t ws_size,
                              hipStream_t stream) {
  (void)n_in; (void)d_ws; (void)ws_size; (void)out_size;
  const float* x = (const float*)d_in[0];
  float* out = (float*)d_out;

  const int rows = in_sizes[0] / ROW;      // 32768 for (1,16,2048,2048)
  if (rows <= 0) return;

  int blocks = rows < BLOCKS ? rows : BLOCKS;
  softmax2048_kernel<<<blocks, TPB, 0, stream>>>(x, out, rows);
}